// TemporalGridEncoder_14817637171836
// MI455X (gfx1250) — compile-verified
//
#include <hip/hip_runtime.h>
#include <cmath>

// ---------------- problem constants (from reference) ----------------
#define TGE_NUM_LEVELS 16
#define TGE_LEVEL_DIM  2
#define TGE_GRID_CH    66            // LEVEL_DIM + TEMPORAL_DIM
#define TGE_OUT_CH     (TGE_NUM_LEVELS * TGE_LEVEL_DIM)   // 32
#define TGE_BLOCK      256
#define TGE_MAX_PARAMS (1 << 17)     // 2**LOG2_HASH

struct LevelParams {
  float    scale;      // f32-rounded per-level scale
  unsigned offset_ch;  // level offset * GRID_CH (pre-multiplied)
  unsigned res_eff;    // resolution + 1 (ALIGN_CORNERS == False)
  unsigned hsize;      // hashmap rows for this level
  unsigned mask;       // hsize-1 (valid when pow2)
  unsigned mode;       // 0 = dense grid (mod is identity), 1 = hash & mask, 2 = hash % hsize
};
struct AllParams { LevelParams lvl[TGE_NUM_LEVELS]; };

// ---------------- CDNA5 async global->LDS path (guarded) ----------------
#if defined(__has_builtin)
#if __has_builtin(__builtin_amdgcn_global_load_async_to_lds_b128) && \
    __has_builtin(__builtin_amdgcn_s_wait_asynccnt)
#define TGE_ASYNC_LDS 1
#endif
#endif

// The builtin expects pointers to a 16-byte int vector, in the global (AS1)
// and LDS (AS3) address spaces respectively (per hipcc's diagnostic).
typedef int tge_v4i __attribute__((vector_size(16)));
typedef __attribute__((address_space(1))) tge_v4i* tge_gv4p;  // global v4i*
typedef __attribute__((address_space(3))) tge_v4i* tge_lv4p;  // LDS v4i*

__global__ __launch_bounds__(TGE_BLOCK) void tge_encode_kernel(
    const float* __restrict__ inputs,   // (B,3)
    const float* __restrict__ tri,      // (B,8)  [wa0,ia0,wb0,ib0, wa1,ia1,wb1,ib1]
    const float* __restrict__ emb,      // (TABLE_ROWS, 66) flat
    float* __restrict__ out,            // (B,32)
    AllParams P, int npoints)
{
  __shared__ __align__(16) float s_in [TGE_BLOCK * 3];
  __shared__ __align__(16) float s_tri[TGE_BLOCK * 8];

  const int tid = threadIdx.x;
  const int p0  = blockIdx.x * TGE_BLOCK;
  const int p   = p0 + tid;
  const bool fullblk = (p0 + TGE_BLOCK) <= npoints;   // uniform per block

  if (fullblk) {
    const float* gtri = tri    + (size_t)p0 * 8;   // 8 KB, 16B-aligned
    const float* gin  = inputs + (size_t)p0 * 3;   // 3 KB, 16B-aligned
#if TGE_ASYNC_LDS
    // Each lane DMAs 16 B; whole block stages 11 KB via ASYNCcnt machinery.
    __builtin_amdgcn_global_load_async_to_lds_b128(
        (tge_gv4p)(gtri + tid * 4), (tge_lv4p)&s_tri[tid * 4], 0, 0);
    __builtin_amdgcn_global_load_async_to_lds_b128(
        (tge_gv4p)(gtri + (tid + TGE_BLOCK) * 4),
        (tge_lv4p)&s_tri[(tid + TGE_BLOCK) * 4], 0, 0);
    if (tid < (TGE_BLOCK * 3) / 4)
      __builtin_amdgcn_global_load_async_to_lds_b128(
          (tge_gv4p)(gin + tid * 4), (tge_lv4p)&s_in[tid * 4], 0, 0);
    __builtin_amdgcn_s_wait_asynccnt(0);
#else
    ((float4*)s_tri)[tid]             = ((const float4*)gtri)[tid];
    ((float4*)s_tri)[tid + TGE_BLOCK] = ((const float4*)gtri)[tid + TGE_BLOCK];
    if (tid < (TGE_BLOCK * 3) / 4)
      ((float4*)s_in)[tid] = ((const float4*)gin)[tid];
#endif
    __syncthreads();
  }
  if (p >= npoints) return;

  float ix, iy, iz, wa0, wb0, wa1, wb1;
  int   ia0, ib0, ia1, ib1;
  if (fullblk) {
    ix  = s_in[tid * 3 + 0]; iy = s_in[tid * 3 + 1]; iz = s_in[tid * 3 + 2];
    wa0 = s_tri[tid * 8 + 0]; ia0 = (int)s_tri[tid * 8 + 1];
    wb0 = s_tri[tid * 8 + 2]; ib0 = (int)s_tri[tid * 8 + 3];
    wa1 = s_tri[tid * 8 + 4]; ia1 = (int)s_tri[tid * 8 + 5];
    wb1 = s_tri[tid * 8 + 6]; ib1 = (int)s_tri[tid * 8 + 7];
  } else {
    const float* ip = inputs + (size_t)p * 3;
    ix = ip[0]; iy = ip[1]; iz = ip[2];
    const float* tp = tri + (size_t)p * 8;
    wa0 = tp[0]; ia0 = (int)tp[1]; wb0 = tp[2]; ib0 = (int)tp[3];
    wa1 = tp[4]; ia1 = (int)tp[5]; wb1 = tp[6]; ib1 = (int)tp[7];
  }

  const LevelParams lp = P.lvl[blockIdx.y];   // kernarg -> SGPRs, wave-uniform

  // pos = in * scale + 0.5 (match reference: separate f32 mul then add)
  const float px = __fadd_rn(__fmul_rn(ix, lp.scale), 0.5f);
  const float py = __fadd_rn(__fmul_rn(iy, lp.scale), 0.5f);
  const float pz = __fadd_rn(__fmul_rn(iz, lp.scale), 0.5f);
  const float flx = floorf(px), fly = floorf(py), flz = floorf(pz);
  const float fx = __fsub_rn(px, flx);
  const float fy = __fsub_rn(py, fly);
  const float fz = __fsub_rn(pz, flz);
  const unsigned x0 = (unsigned)flx, y0 = (unsigned)fly, z0 = (unsigned)flz;

  const float wx[2] = {1.0f - fx, fx};
  const float wy[2] = {1.0f - fy, fy};
  const float wz[2] = {1.0f - fz, fz};

  const unsigned re  = lp.res_eff;
  const unsigned re2 = re * re;

  float acc0 = 0.0f, acc1 = 0.0f;
#pragma unroll
  for (int c = 0; c < 8; ++c) {
    const unsigned ox = (unsigned)(c & 1);
    const unsigned oy = (unsigned)((c >> 1) & 1);
    const unsigned oz = (unsigned)(c >> 2);
    const unsigned cx = x0 + ox, cy = y0 + oy, cz = z0 + oz;

    unsigned idx;
    if (lp.mode == 0u) {
      idx = cx + cy * re + cz * re2;                     // provably < hsize
    } else {
      idx = (cx * 1u) ^ (cy * 2654435761u) ^ (cz * 805459861u);
      idx = (lp.mode == 1u) ? (idx & lp.mask) : (idx % lp.hsize);
    }

    const float w = __fmul_rn(__fmul_rn(wx[ox], wy[oy]), wz[oz]);
    const float* __restrict__ r = emb + (size_t)(idx * TGE_GRID_CH + lp.offset_ch);
    const float va0 = r[ia0], vb0 = r[ib0], va1 = r[ia1], vb1 = r[ib1];
    acc0 = __fadd_rn(acc0,
           __fmul_rn(w, __fadd_rn(__fmul_rn(wa0, va0), __fmul_rn(wb0, vb0))));
    acc1 = __fadd_rn(acc1,
           __fmul_rn(w, __fadd_rn(__fmul_rn(wa1, va1), __fmul_rn(wb1, vb1))));
  }

  float2 o2; o2.x = acc0; o2.y = acc1;
  *(float2*)(out + (size_t)p * TGE_OUT_CH + (size_t)(blockIdx.y * 2)) = o2;
}

// ---------------- host side ----------------
static void tge_build_params(AllParams& P) {
  // Mirrors _level_params() in float64, matching numpy's exp2/log2/ceil path.
  const double S = log2(2048.0 / 16.0) / 15.0;   // log2(DESIRED/BASE)/(L-1)
  long long offset = 0;
  for (int l = 0; l < TGE_NUM_LEVELS; ++l) {
    const double scale = exp2((double)l * S) * 16.0 - 1.0;
    const long long resolution = (long long)ceil(scale) + 1;
    const long long res_eff    = resolution + 1;            // ALIGN_CORNERS=False
    const long long n3 = res_eff * res_eff * res_eff;
    long long n = (n3 < (long long)TGE_MAX_PARAMS) ? n3 : (long long)TGE_MAX_PARAMS;
    n = ((n + 7) / 8) * 8;
    const bool hashed = (n3 > n);
    const bool pow2   = ((n & (n - 1)) == 0);

    LevelParams& lp = P.lvl[l];
    lp.scale     = (float)scale;
    lp.offset_ch = (unsigned)(offset * TGE_GRID_CH);
    lp.res_eff   = (unsigned)res_eff;
    lp.hsize     = (unsigned)n;
    lp.mask      = (unsigned)(n - 1);
    lp.mode      = hashed ? (pow2 ? 1u : 2u) : 0u;
    offset += n;
  }
}

extern "C" void kernel_launch(void* const* d_in, const int* in_sizes, int n_in,
                              void* d_out, int out_size, void* d_ws, size_t ws_size,
                              hipStream_t stream) {
  const float* inputs = (const float*)d_in[0];   // (B,3)
  const float* tri    = (const float*)d_in[1];   // (B,8)
  const float* emb    = (const float*)d_in[2];   // (TABLE_ROWS*66)
  float* out = (float*)d_out;
  const int B = in_sizes[0] / 3;

  AllParams P;
  tge_build_params(P);

  dim3 grid((unsigned)((B + TGE_BLOCK - 1) / TGE_BLOCK), TGE_NUM_LEVELS, 1);
  dim3 block(TGE_BLOCK, 1, 1);
  tge_encode_kernel<<<grid, block, 0, stream>>>(inputs, tri, emb, out, P, B);
}